// RGCN_80848464380531
// MI455X (gfx1250) — compile-verified
//
#include <hip/hip_runtime.h>
#include <hip/hip_bf16.h>

typedef float v2f __attribute__((ext_vector_type(2)));
typedef float v8f __attribute__((ext_vector_type(8)));

// ---------------------------------------------------------------------------
// GEMM: Y[z] = X @ M[z], z in [0,R) -> W_r (out to Yr), z==R -> root (to Yroot)
// X: [N,D] row-major, W: [R,D,H], root: [D,H]. fp32 WMMA 16x16x4.
// Block = 256 threads = 8 waves. Each wave computes a 16(M) x 64(N) tile.
// B tile staged in LDS in K-pair-interleaved layout: sB[kpair][n][2] so a
// WMMA B fragment {W[k][n], W[k+1][n]} is one contiguous ds_load_b64.
// A fragments for a whole 64-K chunk are preloaded into registers so the
// global loads issue as one clause instead of stalling every k-step.
// ---------------------------------------------------------------------------
__global__ __launch_bounds__(256) void rgcn_gemm_wmma(
    const float* __restrict__ X,
    const float* __restrict__ W,
    const float* __restrict__ root,
    float* __restrict__ Yr,
    float* __restrict__ Yroot,
    int N, int D, int H, int R)
{
    __shared__ float sB[32 * 128];   // [kpair 0..31][n 0..63][2]

    const int z = blockIdx.z;
    const float* Wp = (z < R) ? (W + (size_t)z * D * H) : root;
    float*       Yp = (z < R) ? (Yr + (size_t)z * N * H) : Yroot;

    const int col0 = blockIdx.y * 64;
    const int wave = threadIdx.x >> 5;
    const int lane = threadIdx.x & 31;
    const int p    = lane & 15;   // M (for A) / N (for B,C,D) index within tile
    const int hi   = lane >> 4;   // K-half selector for A/B, row+8 for C/D

    const int  mtiles = N >> 4;                  // N % 16 == 0 (N = 20000)
    const int  mt     = blockIdx.x * 8 + wave;
    const bool active = (mt < mtiles);
    const int  m0     = (active ? mt : 0) << 4;
    const size_t rowbase = (size_t)(m0 + p) * D;

    v8f acc0 = {}; v8f acc1 = {}; v8f acc2 = {}; v8f acc3 = {};

    const int kp = threadIdx.x >> 4;         // 0..15 (k-pair row for staging)
    const int n4 = (threadIdx.x & 15) << 2;  // 0..60 (n col, 4-wide)

    for (int k0 = 0; k0 < D; k0 += 64) {
        // ---- preload A fragments for this K-chunk (one load clause) ----
        v2f areg[16];
#pragma unroll
        for (int i = 0; i < 16; ++i)
            areg[i] = *(const v2f*)(&X[rowbase + k0 + 4 * i + 2 * hi]);

        __syncthreads();   // previous chunk's readers are done with sB
        // ---- stage B tile, K-pair interleaved ----
#pragma unroll
        for (int j = 0; j < 2; ++j) {
            const int kpair = kp + 16 * j;   // 0..31
            const float4 r0 = *(const float4*)(&Wp[(size_t)(k0 + 2 * kpair    ) * H + col0 + n4]);
            const float4 r1 = *(const float4*)(&Wp[(size_t)(k0 + 2 * kpair + 1) * H + col0 + n4]);
            float* dp = &sB[kpair * 128 + n4 * 2];
            v2f t;
            t.x = r0.x; t.y = r1.x; ((v2f*)dp)[0] = t;
            t.x = r0.y; t.y = r1.y; ((v2f*)dp)[1] = t;
            t.x = r0.z; t.y = r1.z; ((v2f*)dp)[2] = t;
            t.x = r0.w; t.y = r1.w; ((v2f*)dp)[3] = t;
        }
        __syncthreads();

        // ---- 16 k-steps x 4 col-tiles of WMMA ----
#pragma unroll
        for (int i = 0; i < 16; ++i) {
            // B fragment: lane p half hi -> {W[k+2hi][n], W[k+2hi+1][n]}
            const float* bp = &sB[(2 * i + hi) * 128 + p * 2];
            v2f b;
            b = *(const v2f*)(bp + 0);
            acc0 = __builtin_amdgcn_wmma_f32_16x16x4_f32(false, areg[i], false, b, (short)0, acc0, false, false);
            b = *(const v2f*)(bp + 32);
            acc1 = __builtin_amdgcn_wmma_f32_16x16x4_f32(false, areg[i], false, b, (short)0, acc1, false, false);
            b = *(const v2f*)(bp + 64);
            acc2 = __builtin_amdgcn_wmma_f32_16x16x4_f32(false, areg[i], false, b, (short)0, acc2, false, false);
            b = *(const v2f*)(bp + 96);
            acc3 = __builtin_amdgcn_wmma_f32_16x16x4_f32(false, areg[i], false, b, (short)0, acc3, false, false);
        }
    }

    if (active) {
        // D layout: VGPR j -> row m0 + j + 8*hi, col col0 + 16t + p
#pragma unroll
        for (int j = 0; j < 8; ++j) {
            const size_t r = (size_t)(m0 + j + 8 * hi) * H + col0 + p;
            Yp[r +  0] = acc0[j];
            Yp[r + 16] = acc1[j];
            Yp[r + 32] = acc2[j];
            Yp[r + 48] = acc3[j];
        }
    }
}

// ---------------------------------------------------------------------------
// Edge scatter: one wave per edge; gather xr[(r,src)] (256 f32 via float4)
// and atomic-add into sums[(r,dst)]; lane0 bumps the edge count.
// ---------------------------------------------------------------------------
__global__ __launch_bounds__(256) void rgcn_scatter(
    const float* __restrict__ xr,
    const int* __restrict__ src, const int* __restrict__ dst,
    const int* __restrict__ et,
    float* __restrict__ sums, float* __restrict__ cnt,
    int N, int E, int H)
{
    const int e = blockIdx.x * 8 + (threadIdx.x >> 5);
    if (e >= E) return;
    const int lane = threadIdx.x & 31;
    const int r = et[e];
    const int s = src[e];
    const int d = dst[e];
    const float4* in  = (const float4*)(xr + ((size_t)r * N + s) * H);
    float*        out = sums + ((size_t)r * N + d) * H;
#pragma unroll
    for (int j = 0; j < 2; ++j) {
        const int q = lane + 32 * j;   // H/4 = 64 float4 per edge
        const float4 v = in[q];
        atomicAdd(out + 4 * q + 0, v.x);
        atomicAdd(out + 4 * q + 1, v.y);
        atomicAdd(out + 4 * q + 2, v.z);
        atomicAdd(out + 4 * q + 3, v.w);
    }
    if (lane == 0) atomicAdd(cnt + (size_t)r * N + d, 1.0f);
}

// ---------------------------------------------------------------------------
// Finalize: out[i,h] = relu?( sum_r sums[r,i,h]/max(cnt[r,i],1) + root[i,h] + bias[h] )
// ---------------------------------------------------------------------------
__global__ __launch_bounds__(256) void rgcn_finalize(
    const float* __restrict__ sums, const float* __restrict__ cnt,
    const float* __restrict__ rootb, const float* __restrict__ bias,
    float* __restrict__ out, int N, int H, int R, int do_relu)
{
    const size_t idx = (size_t)blockIdx.x * blockDim.x + threadIdx.x;
    if (idx >= (size_t)N * H) return;
    const int i = (int)(idx / H);
    const int h = (int)(idx % H);
    float acc = rootb[idx] + bias[h];
    for (int r = 0; r < R; ++r) {
        const size_t seg = (size_t)r * N + i;
        acc += sums[seg * H + h] / fmaxf(cnt[seg], 1.0f);
    }
    if (do_relu) acc = fmaxf(acc, 0.0f);
    out[idx] = acc;
}

// ---------------------------------------------------------------------------
// Global add pool: g[h] += sum over a stripe of rows (one column per thread).
// ---------------------------------------------------------------------------
__global__ __launch_bounds__(256) void rgcn_pool(
    const float* __restrict__ hfeat, float* __restrict__ g, int N, int H)
{
    const int h = threadIdx.x;  // H == 256
    float acc = 0.0f;
    for (int i = blockIdx.x; i < N; i += gridDim.x)
        acc += hfeat[(size_t)i * H + h];
    atomicAdd(&g[h], acc);
}

// ---------------------------------------------------------------------------
// Head: out[c] = sum_h g[h] * lin_w[h,c] + lin_b[c], c in {0,1}.
// ---------------------------------------------------------------------------
__global__ __launch_bounds__(256) void rgcn_head(
    const float* __restrict__ g, const float* __restrict__ lw,
    const float* __restrict__ lb, float* __restrict__ out, int H)
{
    __shared__ float s0[256];
    __shared__ float s1[256];
    const int t = threadIdx.x;
    const float gv = (t < H) ? g[t] : 0.0f;
    s0[t] = gv * ((t < H) ? lw[t * 2 + 0] : 0.0f);
    s1[t] = gv * ((t < H) ? lw[t * 2 + 1] : 0.0f);
    __syncthreads();
    for (int stride = 128; stride > 0; stride >>= 1) {
        if (t < stride) { s0[t] += s0[t + stride]; s1[t] += s1[t + stride]; }
        __syncthreads();
    }
    if (t == 0) { out[0] = s0[0] + lb[0]; out[1] = s1[0] + lb[1]; }
}

// ---------------------------------------------------------------------------
extern "C" void kernel_launch(void* const* d_in, const int* in_sizes, int n_in,
                              void* d_out, int out_size, void* d_ws, size_t ws_size,
                              hipStream_t stream)
{
    const float* x     = (const float*)d_in[0];
    const int*   eidx  = (const int*)d_in[1];
    const int*   etype = (const int*)d_in[2];
    // d_in[3] edge_attr: unused by the reference
    const float* W0    = (const float*)d_in[4];
    const float* root0 = (const float*)d_in[5];
    const float* b0    = (const float*)d_in[6];
    const float* W1    = (const float*)d_in[7];
    const float* root1 = (const float*)d_in[8];
    const float* b1    = (const float*)d_in[9];
    const float* linw  = (const float*)d_in[10];
    const float* linb  = (const float*)d_in[11];

    const int H = in_sizes[6];
    const int D = in_sizes[5] / H;
    const int N = in_sizes[0] / D;
    const int E = in_sizes[2];
    const int R = in_sizes[4] / (D * H);

    float* ws = (float*)d_ws;
    size_t off = 0;
    float* xr    = ws + off; off += (size_t)R * N * H;  // relation features
    float* sums  = ws + off; off += (size_t)R * N * H;  // segment sums
    float* cnt   = ws + off; off += (size_t)R * N;      // contiguous after sums
    float* rootb = ws + off; off += (size_t)N * H;      // x @ root
    float* h1    = ws + off; off += (size_t)N * H;      // layer output
    float* g     = ws + off; off += (size_t)H;          // pooled vector

    const int* srcp = eidx;
    const int* dstp = eidx + E;

    const dim3 blk(256);
    const dim3 ggrid((unsigned)((N / 16 + 7) / 8), (unsigned)(H / 64), (unsigned)(R + 1));
    const dim3 sgrid((unsigned)((E + 7) / 8));
    const dim3 fgrid((unsigned)(((size_t)N * H + 255) / 256));
    const size_t zbytes = ((size_t)R * N * H + (size_t)R * N) * sizeof(float);

    // ---- layer 0 ----
    hipMemsetAsync(sums, 0, zbytes, stream);
    rgcn_gemm_wmma<<<ggrid, blk, 0, stream>>>(x, W0, root0, xr, rootb, N, D, H, R);
    rgcn_scatter <<<sgrid, blk, 0, stream>>>(xr, srcp, dstp, etype, sums, cnt, N, E, H);
    rgcn_finalize<<<fgrid, blk, 0, stream>>>(sums, cnt, rootb, b0, h1, N, H, R, 1);

    // ---- layer 1 (input h1; finalize overwrites h1 in place, safe) ----
    hipMemsetAsync(sums, 0, zbytes, stream);
    rgcn_gemm_wmma<<<ggrid, blk, 0, stream>>>(h1, W1, root1, xr, rootb, N, H, H, R);
    rgcn_scatter <<<sgrid, blk, 0, stream>>>(xr, srcp, dstp, etype, sums, cnt, N, E, H);
    rgcn_finalize<<<fgrid, blk, 0, stream>>>(sums, cnt, rootb, b1, h1, N, H, R, 0);

    // ---- pool + head ----
    hipMemsetAsync(g, 0, (size_t)H * sizeof(float), stream);
    rgcn_pool<<<dim3(256), blk, 0, stream>>>(h1, g, N, H);
    rgcn_head<<<dim3(1),   blk, 0, stream>>>(g, linw, linb, (float*)d_out, H);
}